// Capsule_31112743092780
// MI455X (gfx1250) — compile-verified
//
#include <hip/hip_runtime.h>
#include <hip/hip_bf16.h>

#define B_   128
#define C1_  1152
#define D1_  8
#define C2_  10
#define D2_  16
#define JD_  (C2_*D2_)   // 160 = rows of A / (j,d) pairs

typedef __attribute__((ext_vector_type(2))) float v2f;
typedef __attribute__((ext_vector_type(4))) float v4f;
typedef __attribute__((ext_vector_type(8))) float v8f;

// ---------------------------------------------------------------------------
// Kernel 1: PrimaryCaps squash.  prim[b,i,:] = squash(inputs[b, i*8 : i*8+8])
// ---------------------------------------------------------------------------
__global__ void squash_prim_kernel(const float* __restrict__ in,
                                   float* __restrict__ prim) {
  int cap = blockIdx.x * blockDim.x + threadIdx.x;   // b*C1 + i
  if (cap >= B_ * C1_) return;
  const float* p = in + (size_t)cap * D1_;
  v4f a = *(const v4f*)(p);
  v4f b = *(const v4f*)(p + 4);
  float sq = a[0]*a[0] + a[1]*a[1] + a[2]*a[2] + a[3]*a[3]
           + b[0]*b[0] + b[1]*b[1] + b[2]*b[2] + b[3]*b[3];
  float scale = (sq / (1.0f + sq)) * __frsqrt_rn(sq + 1e-8f);
  a *= scale;
  b *= scale;
  float* q = prim + (size_t)cap * D1_;
  *(v4f*)(q)     = a;
  *(v4f*)(q + 4) = b;
}

// ---------------------------------------------------------------------------
// Kernel 2: u_hat[b,i,jd] = sum_k W[i,jd,k] * prim[b,i,k]  via f32 WMMA.
// Per capsule i: A = W[i] (160x8), B = prim[:,i,:]^T (8x128), D = A@B (160x128).
// grid.x = C1 (one block per i), 256 threads = 8 waves; wave w = batch tile w.
// Each 16x16 output tile = two chained V_WMMA_F32_16X16X4_F32 (K=0..3, 4..7).
//
// 32-bit A 16x4 lane layout (ISA 7.12.2): lane%16 = M row, lane/16 selects
// K base {0,2}, VGPR index adds K.  B mirrors it with N = lane%16.
// C/D: lane%16 = N col, rows M = 8*(lane/16) + vgpr -> 8 contiguous floats
// per lane in u_hat's (j,d)-fastest layout => b128 stores.
// ---------------------------------------------------------------------------
__global__ __launch_bounds__(256) void uhat_wmma_kernel(
    const float* __restrict__ W,
    const float* __restrict__ prim,
    float* __restrict__ uhat) {
  int i    = blockIdx.x;             // capsule 0..1151
  int wave = threadIdx.x >> 5;       // 0..7 -> batch tile
  int lane = threadIdx.x & 31;
  int ln   = lane & 15;
  int hi   = lane >> 4;              // 0/1
  int K0   = hi * 2;                 // K base for this half-wave
  int bcol = wave * 16 + ln;         // batch column 0..127

  // B-matrix fragments (prim): rows K, cols = batch
  const float* pb = prim + ((size_t)bcol * C1_ + i) * D1_;
  v2f b_lo = *(const v2f*)(pb + K0);        // K = K0, K0+1
  v2f b_hi = *(const v2f*)(pb + 4 + K0);    // K = 4+K0, 5+K0

  for (int mt = 0; mt < 10; ++mt) {
    int row = mt * 16 + ln;                 // A row = j*16 + d, 0..159
    const float* pa = W + ((size_t)i * JD_ + row) * D1_;
    v2f a_lo = *(const v2f*)(pa + K0);
    v2f a_hi = *(const v2f*)(pa + 4 + K0);

    v8f acc = {};
    acc = __builtin_amdgcn_wmma_f32_16x16x4_f32(
        false, a_lo, false, b_lo, (short)0, acc, false, false);
    acc = __builtin_amdgcn_wmma_f32_16x16x4_f32(
        false, a_hi, false, b_hi, (short)0, acc, false, false);

    // lane holds rows M = hi*8 .. hi*8+7 for column bcol -> contiguous in uhat
    float* o = uhat + (((size_t)bcol * C1_ + i) * JD_ + mt * 16 + hi * 8);
    v4f lo = {acc[0], acc[1], acc[2], acc[3]};
    v4f hiv = {acc[4], acc[5], acc[6], acc[7]};
    *(v4f*)(o)     = lo;
    *(v4f*)(o + 4) = hiv;
  }
}

// ---------------------------------------------------------------------------
// Kernel 3: fused dynamic routing (3 iterations).  One block per batch elem.
// raw[i][j] lives in LDS; u_hat streamed from L2 twice per iteration.
// ---------------------------------------------------------------------------
__global__ __launch_bounds__(512) void routing_kernel(
    const float* __restrict__ uhat,
    const float* __restrict__ bias,
    float* __restrict__ out) {
  __shared__ float raw[C1_ * C2_];     // 46080 B
  __shared__ float rmax[C1_];          //  4608 B
  __shared__ float denom[C1_];         //  4608 B
  __shared__ float sPart[3 * JD_];
  __shared__ float sArr[JD_];
  __shared__ float vArr[JD_];
  __shared__ float scaleArr[C2_];

  int b   = blockIdx.x;
  int tid = threadIdx.x;
  const float* ub = uhat + (size_t)b * C1_ * JD_;

  for (int p = tid; p < C1_ * C2_; p += 512) raw[p] = 0.0f;
  __syncthreads();

  for (int r = 0; r < 3; ++r) {
    // --- per-i softmax max & denominator over the 10 output caps ---
    for (int i = tid; i < C1_; i += 512) {
      float m = raw[i * C2_];
      #pragma unroll
      for (int j = 1; j < C2_; ++j) m = fmaxf(m, raw[i * C2_ + j]);
      float ds = 0.0f;
      #pragma unroll
      for (int j = 0; j < C2_; ++j) ds += __expf(raw[i * C2_ + j] - m);
      rmax[i]  = m;
      denom[i] = ds;
    }
    __syncthreads();

    // --- s[jd] = sum_i c[i,j] * u[b,i,jd]  (3-way split of the C1 sum) ---
    if (tid < 3 * JD_) {
      int g  = tid / JD_;
      int jd = tid % JD_;
      int j  = jd >> 4;
      float acc = 0.0f;
      for (int i = g; i < C1_; i += 3) {
        __builtin_prefetch(&ub[(size_t)(i + 24) * JD_ + jd], 0, 0);
        float c = __expf(raw[i * C2_ + j] - rmax[i]) * __frcp_rn(denom[i]);
        acc = fmaf(c, ub[(size_t)i * JD_ + jd], acc);
      }
      sPart[tid] = acc;
    }
    __syncthreads();

    if (tid < JD_)
      sArr[tid] = sPart[tid] + sPart[JD_ + tid] + sPart[2 * JD_ + tid] + bias[tid];
    __syncthreads();

    // --- squash over D2 ---
    if (tid < C2_) {
      float sq = 0.0f;
      #pragma unroll
      for (int d = 0; d < D2_; ++d) {
        float x = sArr[tid * D2_ + d];
        sq = fmaf(x, x, sq);
      }
      scaleArr[tid] = (sq / (1.0f + sq)) * __frsqrt_rn(sq + 1e-8f);
    }
    __syncthreads();

    if (tid < JD_) {
      float v = sArr[tid] * scaleArr[tid >> 4];
      vArr[tid] = v;
      if (r == 2) out[(size_t)b * JD_ + tid] = v;   // [B,1,C2,D2,1]
    }
    __syncthreads();

    // --- agreement: raw[i][j] += <u[b,i,j,:], v[j,:]> ---
    if (r < 2) {
      for (int p = tid; p < C1_ * C2_; p += 512) {
        int i = p / C2_;
        int j = p % C2_;
        const float* up = ub + (size_t)i * JD_ + j * D2_;
        v4f u0 = *(const v4f*)(up);
        v4f u1 = *(const v4f*)(up + 4);
        v4f u2 = *(const v4f*)(up + 8);
        v4f u3 = *(const v4f*)(up + 12);
        const float* vp = vArr + j * D2_;
        float dot = 0.0f;
        #pragma unroll
        for (int d = 0; d < 4; ++d) dot = fmaf(u0[d], vp[d],      dot);
        #pragma unroll
        for (int d = 0; d < 4; ++d) dot = fmaf(u1[d], vp[4 + d],  dot);
        #pragma unroll
        for (int d = 0; d < 4; ++d) dot = fmaf(u2[d], vp[8 + d],  dot);
        #pragma unroll
        for (int d = 0; d < 4; ++d) dot = fmaf(u3[d], vp[12 + d], dot);
        raw[p] += dot;
      }
      __syncthreads();
    }
  }
}

// ---------------------------------------------------------------------------
extern "C" void kernel_launch(void* const* d_in, const int* in_sizes, int n_in,
                              void* d_out, int out_size, void* d_ws, size_t ws_size,
                              hipStream_t stream) {
  (void)in_sizes; (void)n_in; (void)out_size; (void)ws_size;
  const float* inputs = (const float*)d_in[0];   // [B, C1*D1]
  const float* W      = (const float*)d_in[1];   // [C1, C2, D2, D1]
  const float* bias   = (const float*)d_in[2];   // [1,1,C2,D2] = 160
  float* out = (float*)d_out;                    // [B,1,C2,D2,1] = B*160

  float* prim = (float*)d_ws;                          // B*C1*D1 floats
  float* uhat = prim + (size_t)B_ * C1_ * D1_;         // B*C1*C2*D2 floats

  // 1) squash primary capsules
  {
    int n = B_ * C1_;
    dim3 grid((n + 255) / 256), block(256);
    squash_prim_kernel<<<grid, block, 0, stream>>>(inputs, prim);
  }
  // 2) u_hat via f32 WMMA GEMMs (one block per input capsule)
  {
    dim3 grid(C1_), block(256);
    uhat_wmma_kernel<<<grid, block, 0, stream>>>(W, prim, uhat);
  }
  // 3) fused dynamic routing (one block per batch element)
  {
    dim3 grid(B_), block(512);
    routing_kernel<<<grid, block, 0, stream>>>(uhat, bias, out);
  }
}